// KernelSparseAttentionLayer_54391465836690
// MI455X (gfx1250) — compile-verified
//
#include <hip/hip_runtime.h>
#include <hip/hip_bf16.h>
#include <math.h>

typedef __attribute__((ext_vector_type(2))) float v2f;
typedef __attribute__((ext_vector_type(8))) float v8f;

#define NTOK 2048
#define DDIM 32
#define RDIM 64
#define JSPLIT 8
#define LOG2E 1.4426950408889634f
#define INV_SQRT_D 0.17677669529663687f /* 1/sqrt(32) */

// ---------------------------------------------------------------------------
// Kernel 1: per-row Q,K,V projections, row norms, random features phi.
// One wave per row; lane = output column. Q is stored pre-scaled by
// (1/sqrt(d))*log2(e) so the N^2 kernel only needs v_exp (exp2) directly.
// ---------------------------------------------------------------------------
__global__ void prep_kernel(const float* __restrict__ Z,
                            const float* __restrict__ Wq, const float* __restrict__ bq,
                            const float* __restrict__ Wk, const float* __restrict__ bk,
                            const float* __restrict__ Wv, const float* __restrict__ bv,
                            const float* __restrict__ omega,
                            float* __restrict__ Qs, float* __restrict__ K,
                            float* __restrict__ V,
                            float* __restrict__ phiQ, float* __restrict__ phiK) {
  const int gid  = blockIdx.x * blockDim.x + threadIdx.x;
  const int row  = gid >> 5;
  const int lane = gid & 31;
  if (row >= NTOK) return;

  float z = Z[row * DDIM + lane];
  float q = bq[lane], k = bk[lane], v = bv[lane];
#pragma unroll
  for (int kk = 0; kk < DDIM; ++kk) {
    float zk = __shfl(z, kk, 32);
    q = fmaf(zk, Wq[kk * DDIM + lane], q);
    k = fmaf(zk, Wk[kk * DDIM + lane], k);
    v = fmaf(zk, Wv[kk * DDIM + lane], v);
  }
  K[row * DDIM + lane] = k;
  V[row * DDIM + lane] = v;
  Qs[row * DDIM + lane] = q * (INV_SQRT_D * LOG2E);

  float sq = q * q, sk = k * k;
#pragma unroll
  for (int m = 16; m >= 1; m >>= 1) {
    sq += __shfl_xor(sq, m, 32);
    sk += __shfl_xor(sk, m, 32);
  }
  float qn = q / fmaxf(sqrtf(sq), 1e-6f);
  float kn = k / fmaxf(sqrtf(sk), 1e-6f);

  float pq0 = 0.f, pq1 = 0.f, pk0 = 0.f, pk1 = 0.f;
#pragma unroll
  for (int kk = 0; kk < DDIM; ++kk) {
    float a = __shfl(qn, kk, 32);
    float b = __shfl(kn, kk, 32);
    const float* om = omega + kk * RDIM;
    pq0 = fmaf(a, om[lane], pq0);
    pq1 = fmaf(a, om[lane + 32], pq1);
    pk0 = fmaf(b, om[lane], pk0);
    pk1 = fmaf(b, om[lane + 32], pk1);
  }
  const float inv_sqrt_r = 0.125f;
  phiQ[row * RDIM + lane]      = __builtin_amdgcn_exp2f(pq0 * LOG2E) * inv_sqrt_r;
  phiQ[row * RDIM + lane + 32] = __builtin_amdgcn_exp2f(pq1 * LOG2E) * inv_sqrt_r;
  phiK[row * RDIM + lane]      = __builtin_amdgcn_exp2f(pk0 * LOG2E) * inv_sqrt_r;
  phiK[row * RDIM + lane + 32] = __builtin_amdgcn_exp2f(pk1 * LOG2E) * inv_sqrt_r;
}

// ---------------------------------------------------------------------------
// Kernel 2: KV = phi_K^T @ V  [64 x 32]  (tiny reduction, data resident in L2)
// ---------------------------------------------------------------------------
__global__ void kv_kernel(const float* __restrict__ phiK, const float* __restrict__ V,
                          float* __restrict__ KV) {
  const int t = blockIdx.x * blockDim.x + threadIdx.x; // 0..2047
  const int r = t >> 5, c = t & 31;
  float s = 0.f;
  for (int i = 0; i < NTOK; ++i)
    s = fmaf(phiK[i * RDIM + r], V[i * DDIM + c], s);
  KV[r * DDIM + c] = s;
}

// ---------------------------------------------------------------------------
// Kernel 3: attn base = low_rank = phi_Q @ KV  [N x 32]
// ---------------------------------------------------------------------------
__global__ void lowrank_kernel(const float* __restrict__ phiQ, const float* __restrict__ KV,
                               float* __restrict__ attn) {
  const int t = blockIdx.x * blockDim.x + threadIdx.x; // 0..65535
  const int i = t >> 5, c = t & 31;
  float s = 0.f;
#pragma unroll 8
  for (int r = 0; r < RDIM; ++r)
    s = fmaf(phiQ[i * RDIM + r], KV[r * DDIM + c], s);
  attn[t] = s;
}

// ---------------------------------------------------------------------------
// Kernel 4 (hot): per (16-row block, j-chunk) wave.
// Per 16x16 tile, all in the WMMA C/D layout (lane l: rows v+8*(l/16),
// col l%16):
//   P    = phi_Q_tile @ phi_K_tile^T      -> 16x V_WMMA_F32_16X16X4_F32
//   e    = sum_k exp2(Qs[i,k]*K[j,k])     -> VALU/trans (co-executes w/ WMMA)
//   S    = mask==0 ? e - P : 0
//   S -> LDS transpose -> A-operand layout
//   partial += S(16x16) @ V_tile(16x32)   -> 8x V_WMMA_F32_16X16X4_F32
// ---------------------------------------------------------------------------
__global__ __launch_bounds__(32) void score_kernel(
    const float* __restrict__ Qs, const float* __restrict__ Kg,
    const float* __restrict__ Vg, const float* __restrict__ phiQ,
    const float* __restrict__ phiK, const int* __restrict__ mask,
    float* __restrict__ part) {
  __shared__ __align__(16) float Ql[16][36];   // pads chosen bank-conflict-free
  __shared__ __align__(16) float pQl[16][68];
  __shared__ __align__(16) float Kl[16][36];
  __shared__ __align__(16) float pKl[16][68];
  __shared__ __align__(16) float Vl[16][36];
  __shared__ __align__(16) float Sl[16][20];   // score-tile transpose buffer

  const int lane = threadIdx.x;
  const int hi = lane >> 4;      // 0: lanes 0-15, 1: lanes 16-31
  const int il = lane & 15;
  const int i0 = blockIdx.x * 16;
  const int jchunk = blockIdx.y;

  // stage the Q block (pre-scaled) and phiQ block once
  for (int e = lane; e < 16 * 8; e += 32) {
    int r = e >> 3, c = e & 7;
    *(float4*)&Ql[r][c * 4] = *(const float4*)&Qs[(i0 + r) * DDIM + c * 4];
  }
  for (int e = lane; e < 16 * 16; e += 32) {
    int r = e >> 4, c = e & 15;
    *(float4*)&pQl[r][c * 4] = *(const float4*)&phiQ[(i0 + r) * RDIM + c * 4];
  }

  v8f acc0 = {};
  v8f acc1 = {};

  const int jbeg = jchunk * (NTOK / JSPLIT);
  const int jend = jbeg + (NTOK / JSPLIT);
  for (int j0 = jbeg; j0 < jend; j0 += 16) {
    // stage K / V / phiK tiles for columns j0..j0+15
    for (int e = lane; e < 16 * 8; e += 32) {
      int r = e >> 3, c = e & 7;
      *(float4*)&Kl[r][c * 4] = *(const float4*)&Kg[(j0 + r) * DDIM + c * 4];
      *(float4*)&Vl[r][c * 4] = *(const float4*)&Vg[(j0 + r) * DDIM + c * 4];
    }
    for (int e = lane; e < 16 * 16; e += 32) {
      int r = e >> 4, c = e & 15;
      *(float4*)&pKl[r][c * 4] = *(const float4*)&phiK[(j0 + r) * RDIM + c * 4];
    }
    __syncthreads();

    // phi_term P = phiQ(16x64) @ phiK^T(64x16) via 16 chained fp32 WMMAs.
    // A: M=il, K=4c+p+2*hi -> pQl[il][...]; B: N=il, K=r -> pKl[il][...]
    v8f P = {};
#pragma unroll
    for (int c = 0; c < 16; ++c) {
      const int rb = 4 * c + 2 * hi;
      v2f a; a.x = pQl[il][rb]; a.y = pQl[il][rb + 1];
      v2f b; b.x = pKl[il][rb]; b.y = pKl[il][rb + 1];
      P = __builtin_amdgcn_wmma_f32_16x16x4_f32(false, a, false, b,
                                                (short)0, P, false, false);
    }

    // exp_term in D layout: lane covers col j=il, rows 8*hi+v
    float ev[8];
#pragma unroll
    for (int v = 0; v < 8; ++v) ev[v] = 0.f;
#pragma unroll 8
    for (int k = 0; k < DDIM; ++k) {
      float kk = Kl[il][k];
#pragma unroll
      for (int v = 0; v < 8; ++v)
        ev[v] += __builtin_amdgcn_exp2f(Ql[8 * hi + v][k] * kk);
    }

    // S = attendable ? e - P : 0   (coalesced mask rows)
    float s[8];
#pragma unroll
    for (int v = 0; v < 8; ++v) {
      int m = mask[(size_t)(i0 + 8 * hi + v) * NTOK + j0 + il];
      float sv = ev[v] - P[v];
      s[v] = (m != 0) ? 0.f : sv;
    }

    // transpose S through LDS into A-operand layout
#pragma unroll
    for (int v = 0; v < 8; ++v) Sl[8 * hi + v][il] = s[v];
    __syncthreads();

    // partial += S(16x16) @ V(16x32): 4 K-steps x 2 N-tiles
#pragma unroll
    for (int c = 0; c < 4; ++c) {
      const int kb = 4 * c + 2 * hi;
      v2f a; a.x = Sl[il][kb]; a.y = Sl[il][kb + 1];
      v2f b0, b1;
      b0.x = Vl[kb][il];      b0.y = Vl[kb + 1][il];
      b1.x = Vl[kb][il + 16]; b1.y = Vl[kb + 1][il + 16];
      acc0 = __builtin_amdgcn_wmma_f32_16x16x4_f32(false, a, false, b0,
                                                   (short)0, acc0, false, false);
      acc1 = __builtin_amdgcn_wmma_f32_16x16x4_f32(false, a, false, b1,
                                                   (short)0, acc1, false, false);
    }
    __syncthreads();
  }

  // write partial tile (C/D layout: elem v -> row v+8*hi, col il(+16))
  float* p = part + (size_t)jchunk * NTOK * DDIM;
#pragma unroll
  for (int v = 0; v < 8; ++v) {
    p[(i0 + v + 8 * hi) * DDIM + il]      = acc0[v];
    p[(i0 + v + 8 * hi) * DDIM + il + 16] = acc1[v];
  }
}

// ---------------------------------------------------------------------------
// Kernel 5: deterministic fixed-order reduction of j-chunk partials onto
// the low_rank base.
// ---------------------------------------------------------------------------
__global__ void reduce_kernel(float* __restrict__ attn, const float* __restrict__ part) {
  const int t = blockIdx.x * blockDim.x + threadIdx.x; // 0..65535
  float s = attn[t];
#pragma unroll
  for (int jc = 0; jc < JSPLIT; ++jc)
    s += part[(size_t)jc * NTOK * DDIM + t];
  attn[t] = s;
}

// ---------------------------------------------------------------------------
// Kernel 6: denominator normalize, Wo projection, residual+LN, FFN, LN.
// One wave per row, all reductions/broadcasts via wave32 shuffles.
// ---------------------------------------------------------------------------
__global__ void final_kernel(const float* __restrict__ Z, const float* __restrict__ attn,
                             const float* __restrict__ Wo, const float* __restrict__ bo,
                             const float* __restrict__ W1, const float* __restrict__ b1,
                             const float* __restrict__ W2, const float* __restrict__ b2,
                             const float* __restrict__ g1, const float* __restrict__ beta1,
                             const float* __restrict__ g2, const float* __restrict__ beta2,
                             float* __restrict__ out) {
  const int gid  = blockIdx.x * blockDim.x + threadIdx.x;
  const int row  = gid >> 5;
  const int lane = gid & 31;
  if (row >= NTOK) return;

  float a = attn[row * DDIM + lane];
  float denom = a;
#pragma unroll
  for (int m = 16; m >= 1; m >>= 1) denom += __shfl_xor(denom, m, 32);
  a /= fmaxf(denom, 1e-6f);

  float o = bo[lane];
#pragma unroll
  for (int kk = 0; kk < DDIM; ++kk)
    o = fmaf(__shfl(a, kk, 32), Wo[kk * DDIM + lane], o);

  float z1 = Z[row * DDIM + lane] + o;
  // LayerNorm 1
  float mu = z1;
#pragma unroll
  for (int m = 16; m >= 1; m >>= 1) mu += __shfl_xor(mu, m, 32);
  mu *= (1.0f / 32.0f);
  float dv = z1 - mu;
  float var = dv * dv;
#pragma unroll
  for (int m = 16; m >= 1; m >>= 1) var += __shfl_xor(var, m, 32);
  var *= (1.0f / 32.0f);
  float z1n = dv * rsqrtf(var + 1e-5f) * g1[lane] + beta1[lane];

  // FFN: hidden 128 = 4 per lane (column lane + 32u)
  float h[4];
#pragma unroll
  for (int u = 0; u < 4; ++u) h[u] = b1[lane + 32 * u];
#pragma unroll
  for (int kk = 0; kk < DDIM; ++kk) {
    float zk = __shfl(z1n, kk, 32);
    const float* w = W1 + kk * 128;
#pragma unroll
    for (int u = 0; u < 4; ++u) h[u] = fmaf(zk, w[lane + 32 * u], h[u]);
  }
#pragma unroll
  for (int u = 0; u < 4; ++u) h[u] = fmaxf(h[u], 0.f);

  float f = b2[lane];
#pragma unroll
  for (int jj = 0; jj < 128; ++jj) {
    float hv = __shfl(h[jj >> 5], jj & 31, 32);
    f = fmaf(hv, W2[jj * DDIM + lane], f);
  }

  float z2 = z1n + f;
  // LayerNorm 2
  float mu2 = z2;
#pragma unroll
  for (int m = 16; m >= 1; m >>= 1) mu2 += __shfl_xor(mu2, m, 32);
  mu2 *= (1.0f / 32.0f);
  float d2 = z2 - mu2;
  float var2 = d2 * d2;
#pragma unroll
  for (int m = 16; m >= 1; m >>= 1) var2 += __shfl_xor(var2, m, 32);
  var2 *= (1.0f / 32.0f);
  out[row * DDIM + lane] = d2 * rsqrtf(var2 + 1e-5f) * g2[lane] + beta2[lane];
}

// ---------------------------------------------------------------------------
extern "C" void kernel_launch(void* const* d_in, const int* in_sizes, int n_in,
                              void* d_out, int out_size, void* d_ws, size_t ws_size,
                              hipStream_t stream) {
  (void)in_sizes; (void)n_in; (void)out_size; (void)ws_size;
  const float* Z     = (const float*)d_in[0];
  const int*   mask  = (const int*)d_in[1];
  const float* Wq    = (const float*)d_in[2];
  const float* bq    = (const float*)d_in[3];
  const float* Wk    = (const float*)d_in[4];
  const float* bk    = (const float*)d_in[5];
  const float* Wv    = (const float*)d_in[6];
  const float* bv    = (const float*)d_in[7];
  const float* Wo    = (const float*)d_in[8];
  const float* bo    = (const float*)d_in[9];
  const float* W1    = (const float*)d_in[10];
  const float* b1    = (const float*)d_in[11];
  const float* W2    = (const float*)d_in[12];
  const float* b2    = (const float*)d_in[13];
  const float* g1    = (const float*)d_in[14];
  const float* beta1 = (const float*)d_in[15];
  const float* g2    = (const float*)d_in[16];
  const float* beta2 = (const float*)d_in[17];
  const float* omega = (const float*)d_in[18];

  float* ws = (float*)d_ws;
  float* Qs   = ws;                       // 2048*32
  float* K    = Qs + NTOK * DDIM;         // 2048*32
  float* V    = K + NTOK * DDIM;          // 2048*32
  float* phiQ = V + NTOK * DDIM;          // 2048*64
  float* phiK = phiQ + NTOK * RDIM;       // 2048*64
  float* KV   = phiK + NTOK * RDIM;       // 64*32
  float* attn = KV + RDIM * DDIM;         // 2048*32
  float* part = attn + NTOK * DDIM;       // JSPLIT*2048*32

  prep_kernel<<<NTOK / 4, 128, 0, stream>>>(Z, Wq, bq, Wk, bk, Wv, bv, omega,
                                            Qs, K, V, phiQ, phiK);
  kv_kernel<<<(RDIM * DDIM) / 256, 256, 0, stream>>>(phiK, V, KV);
  lowrank_kernel<<<(NTOK * DDIM) / 256, 256, 0, stream>>>(phiQ, KV, attn);
  dim3 sg(NTOK / 16, JSPLIT);
  score_kernel<<<sg, 32, 0, stream>>>(Qs, K, V, phiQ, phiK, mask, part);
  reduce_kernel<<<(NTOK * DDIM) / 256, 256, 0, stream>>>(attn, part);
  final_kernel<<<NTOK / 4, 128, 0, stream>>>(Z, attn, Wo, bo, W1, b1, W2, b2,
                                             g1, beta1, g2, beta2, (float*)d_out);
}